// CPSN_24670292148252
// MI455X (gfx1250) — compile-verified
//
#include <hip/hip_runtime.h>
#include <hip/hip_bf16.h>

// ---------------------------------------------------------------------------
// Problem constants (b=1)
// ---------------------------------------------------------------------------
#define S_IMG   25          // support images (f1)
#define Q_IMG   30          // query images (f2)
#define C_DIM   512
#define CP_DIM  (C_DIM / 2) // channel pairs (packed float2)
#define HW      361         // 19*19
#define HWP     384         // padded
#define TEMP    64
#define NPAIR   (S_IMG * Q_IMG)   // 750
#define NTX     12                 // 384/32 x-tiles (32 wide)
#define NTY     6                  // 384/64 y-tiles (64 wide)
#define NTILES  (NTX * NTY)        // 72 -> 9 per wave (8 waves)

typedef float v2f __attribute__((ext_vector_type(2)));
typedef float v8f __attribute__((ext_vector_type(8)));

// ---------------------------------------------------------------------------
// f32 dense WMMA: D(16x16) = A(16x4) * B(4x16) + C   (V_WMMA_F32_16X16X4_F32)
// A/B per ISA layout: lanes 0-15 -> K={0,1}, lanes 16-31 -> K={2,3}
// C/D: vgpr v, lane l -> element (M = v + 8*(l>=16), N = l&15)
// ---------------------------------------------------------------------------
__device__ __forceinline__ v8f wmma4(v2f a, v2f b, v8f c) {
    return __builtin_amdgcn_wmma_f32_16x16x4_f32(
        /*neg_a=*/false, a, /*neg_b=*/false, b,
        /*c_mod=*/(short)0, c, /*reuse_a=*/false, /*reuse_b=*/false);
}

// ---------------------------------------------------------------------------
// Monotonic float key + packed (value,index) for max/argmax.
// Tie-break: equal value -> smaller index wins (matches jnp.argmax first-hit),
// via storing (0xFFFFFFFF - idx) in the low dword. Out-of-range candidates
// are fed -inf (key 0x007FFFFF): any finite candidate beats them (branch-free).
// ---------------------------------------------------------------------------
__device__ __forceinline__ unsigned keyf(float f) {
    unsigned u = __float_as_uint(f);
    return (u & 0x80000000u) ? ~u : (u | 0x80000000u);
}
__device__ __forceinline__ float unkeyf(unsigned k) {
    unsigned u = (k & 0x80000000u) ? (k ^ 0x80000000u) : ~k;
    return __uint_as_float(u);
}
__device__ __forceinline__ unsigned long long packvi(float v, int idx) {
    return ((unsigned long long)keyf(v) << 32) |
           (unsigned long long)(0xFFFFFFFFu - (unsigned)idx);
}
__device__ __forceinline__ unsigned long long maxu64(unsigned long long a,
                                                     unsigned long long b) {
    return a > b ? a : b;
}
__device__ __forceinline__ unsigned long long shflxor_u64(unsigned long long v,
                                                          int mask) {
    int lo = __shfl_xor((int)(unsigned)(v & 0xFFFFFFFFull), mask, 32);
    int hi = __shfl_xor((int)(unsigned)(v >> 32), mask, 32);
    return ((unsigned long long)(unsigned)hi << 32) | (unsigned)lo;
}
#define NINF (__int_as_float(0xFF800000))

// ---------------------------------------------------------------------------
// Kernel 1: per-pixel L2 normalize over channels, pad hw 361 -> 384 with
// zeros, and pack channel pairs: out[cp][pix] = {x[2cp][pix], x[2cp+1][pix]}
// so k_sim fragments load as one global_load_b64.
// ---------------------------------------------------------------------------
__global__ __launch_bounds__(128)
void k_normalize(const float* __restrict__ f1, const float* __restrict__ f2,
                 v2f* __restrict__ f1p, v2f* __restrict__ f2p) {
    int img = blockIdx.x / 3;
    int pix = (blockIdx.x % 3) * 128 + threadIdx.x;   // 0..383
    const float* src;
    v2f* dst;
    if (img < Q_IMG) {
        src = f2 + (size_t)img * C_DIM * HW;
        dst = f2p + (size_t)img * CP_DIM * HWP;
    } else {
        int i = img - Q_IMG;
        src = f1 + (size_t)i * C_DIM * HW;
        dst = f1p + (size_t)i * CP_DIM * HWP;
    }
    if (pix >= HW) {                    // zero padding columns
        v2f z = {0.0f, 0.0f};
        for (int cp = 0; cp < CP_DIM; ++cp) dst[(size_t)cp * HWP + pix] = z;
        return;
    }
    float ssq = 0.0f;
    for (int c = 0; c < C_DIM; ++c) {
        float v = src[(size_t)c * HW + pix];
        ssq = fmaf(v, v, ssq);
    }
    float inv = 1.0f / fmaxf(sqrtf(ssq), 1e-12f);
    for (int cp = 0; cp < CP_DIM; ++cp) {
        float v0 = src[(size_t)(2 * cp) * HW + pix];
        float v1 = src[(size_t)(2 * cp + 1) * HW + pix];
        v2f r = {v0 * inv, v1 * inv};
        dst[(size_t)cp * HWP + pix] = r;
    }
}

// ---------------------------------------------------------------------------
// Kernel 2: meta-learner (1x1 conv 512->64, BN, ReLU, 1x1 conv 64->1, BN, ReLU)
// a1: [25][361], a2: [30][361]
// ---------------------------------------------------------------------------
__global__ __launch_bounds__(128)
void k_meta(const float* __restrict__ f1, const float* __restrict__ f2,
            const float* __restrict__ W1, const float* __restrict__ g1,
            const float* __restrict__ b1, const float* __restrict__ m1,
            const float* __restrict__ v1, const float* __restrict__ W2,
            const float* __restrict__ g2, const float* __restrict__ b2,
            const float* __restrict__ m2, const float* __restrict__ v2,
            float* __restrict__ a1, float* __restrict__ a2) {
    int img = blockIdx.x / 3;
    int pix = (blockIdx.x % 3) * 128 + threadIdx.x;
    if (pix >= HW) return;
    const float* src;
    float* dst;
    if (img < Q_IMG) {
        src = f2 + (size_t)img * C_DIM * HW;
        dst = a2 + (size_t)img * HW;
    } else {
        int i = img - Q_IMG;
        src = f1 + (size_t)i * C_DIM * HW;
        dst = a1 + (size_t)i * HW;
    }
    float acc2 = 0.0f;
    for (int pass = 0; pass < 8; ++pass) {
        float acc[8];
#pragma unroll
        for (int j = 0; j < 8; ++j) acc[j] = 0.0f;
        const float* w1p = W1 + (size_t)(pass * 8) * C_DIM;
        for (int c = 0; c < C_DIM; ++c) {
            float xv = src[(size_t)c * HW + pix];
#pragma unroll
            for (int j = 0; j < 8; ++j)
                acc[j] = fmaf(w1p[(size_t)j * C_DIM + c], xv, acc[j]);
        }
#pragma unroll
        for (int j = 0; j < 8; ++j) {
            int t = pass * 8 + j;
            float inv = g1[t] / sqrtf(v1[t] + 1e-5f);
            float bet = b1[t] - m1[t] * inv;
            float yv = fmaxf(fmaf(acc[j], inv, bet), 0.0f);
            acc2 = fmaf(W2[t], yv, acc2);
        }
    }
    float inv2 = g2[0] / sqrtf(v2[0] + 1e-5f);
    float bet2 = b2[0] - m2[0] * inv2;
    dst[pix] = fmaxf(fmaf(acc2, inv2, bet2), 0.0f);
}

// ---------------------------------------------------------------------------
// Kernel 3: per (q,s) pair: 361x361 cosine GEMM via v_wmma_f32_16x16x4_f32,
// fused row/col max+argmax (packed u64, ds_max_u64), fused gather + mean.
// Per-wave macro-tile: 32 (x) x 64 (y) = 2x4 WMMA sub-tiles, 8 accumulators.
// ---------------------------------------------------------------------------
__device__ __forceinline__
void reduce_tile(v8f acc, int X0, int Y0, int m, bool hi,
                 unsigned long long* s12p, unsigned long long* s21p) {
    const int y = Y0 + m;
    const bool yok = y < HW;
    const int xoff = X0 + (hi ? 8 : 0);
    // ---- s12[y] = max over x of C[x,y]; idx = query_index (an x) ----
    unsigned long long pcol = 0ull;
#pragma unroll
    for (int v = 0; v < 8; ++v) {
        int x = xoff + v;
        float vv = (x < HW) ? acc[v] : NINF;     // cndmask, no branch
        pcol = maxu64(pcol, packvi(vv, x));
    }
    pcol = maxu64(pcol, shflxor_u64(pcol, 16));  // join the two M-halves
    if (!hi && yok) atomicMax(&s12p[y], pcol);
    // ---- s21[x] = max over y of C[x,y]; idx = support_index (a y) ----
#pragma unroll
    for (int v = 0; v < 8; ++v) {
        float vv = yok ? acc[v] : NINF;          // cndmask, no branch
        unsigned long long p = packvi(vv, y);
        p = maxu64(p, shflxor_u64(p, 8));
        p = maxu64(p, shflxor_u64(p, 4));
        p = maxu64(p, shflxor_u64(p, 2));
        p = maxu64(p, shflxor_u64(p, 1));
        int x = xoff + v;
        if (m == 0 && x < HW) atomicMax(&s21p[x], p);
    }
}

__global__ __launch_bounds__(256)
void k_sim(const v2f* __restrict__ f1p, const v2f* __restrict__ f2p,
           const float* __restrict__ a1, const float* __restrict__ a2,
           float* __restrict__ out) {
    __shared__ unsigned long long s12p[HWP];   // per support-pixel y
    __shared__ unsigned long long s21p[HWP];   // per query-pixel x
    __shared__ float r1[256], r2[256];

    const int tid  = threadIdx.x;
    const int pair = blockIdx.x;          // = qq*25 + ss  (row order (q,s))
    const int qq = pair / S_IMG;
    const int ss = pair % S_IMG;

    for (int i = tid; i < HWP; i += 256) { s12p[i] = 0ull; s21p[i] = 0ull; }
    __syncthreads();

    const int lane = tid & 31;
    const int wave = tid >> 5;
    const int m    = lane & 15;
    const bool hi  = lane >= 16;
    const int hio  = hi ? 1 : 0;          // channel-pair offset for this half

    // packed layout: img base + cp*HWP + pix, each element = channels {2cp,2cp+1}
    const v2f* Aimg = f2p + (size_t)qq * CP_DIM * HWP + (size_t)hio * HWP;
    const v2f* Bimg = f1p + (size_t)ss * CP_DIM * HWP + (size_t)hio * HWP;

    for (int t = wave; t < NTILES; t += 8) {              // 72/8 = 9 each
        const int X0 = (t / NTY) * 32;
        const int Y0 = (t % NTY) * 64;
        v8f c00 = {}, c01 = {}, c02 = {}, c03 = {};
        v8f c10 = {}, c11 = {}, c12 = {}, c13 = {};
        const v2f* Ab = Aimg + X0 + m;
        const v2f* Bb = Bimg + Y0 + m;
#pragma unroll 4
        for (int k = 0; k < C_DIM; k += 4) {              // cp advances by 2
            const v2f* Ak = Ab + (size_t)(k >> 1) * HWP;
            const v2f* Bk = Bb + (size_t)(k >> 1) * HWP;
            v2f aA0 = Ak[0];        // rows X0..X0+15   (one global_load_b64)
            v2f aA1 = Ak[16];       // rows X0+16..X0+31
            v2f bB0 = Bk[0];        // cols Y0..Y0+15
            v2f bB1 = Bk[16];       // cols Y0+16..Y0+31
            v2f bB2 = Bk[32];       // cols Y0+32..Y0+47
            v2f bB3 = Bk[48];       // cols Y0+48..Y0+63
            c00 = wmma4(aA0, bB0, c00);
            c01 = wmma4(aA0, bB1, c01);
            c02 = wmma4(aA0, bB2, c02);
            c03 = wmma4(aA0, bB3, c03);
            c10 = wmma4(aA1, bB0, c10);
            c11 = wmma4(aA1, bB1, c11);
            c12 = wmma4(aA1, bB2, c12);
            c13 = wmma4(aA1, bB3, c13);
        }
        reduce_tile(c00, X0,      Y0,      m, hi, s12p, s21p);
        reduce_tile(c01, X0,      Y0 + 16, m, hi, s12p, s21p);
        reduce_tile(c02, X0,      Y0 + 32, m, hi, s12p, s21p);
        reduce_tile(c03, X0,      Y0 + 48, m, hi, s12p, s21p);
        reduce_tile(c10, X0 + 16, Y0,      m, hi, s12p, s21p);
        reduce_tile(c11, X0 + 16, Y0 + 16, m, hi, s12p, s21p);
        reduce_tile(c12, X0 + 16, Y0 + 32, m, hi, s12p, s21p);
        reduce_tile(c13, X0 + 16, Y0 + 48, m, hi, s12p, s21p);
    }
    __syncthreads();

    // Epilogue: weight[k] = a1[pair/30][support_index[k]] * a2[qq][query_index[k]]
    // s1 = mean_k(s12[k]*w[k]); s2 = mean_k(s21[k]*w[k])
    const int sprime = pair / Q_IMG;      // faithful (b,s,q)-vs-(b,q,s) quirk
    float p1 = 0.0f, p2 = 0.0f;
    for (int k = tid; k < HW; k += 256) {
        unsigned long long u12 = s12p[k];
        unsigned long long u21 = s21p[k];
        float v12 = unkeyf((unsigned)(u12 >> 32));
        float v21 = unkeyf((unsigned)(u21 >> 32));
        unsigned qIdx = 0xFFFFFFFFu - (unsigned)(u12 & 0xFFFFFFFFull);
        unsigned sIdx = 0xFFFFFFFFu - (unsigned)(u21 & 0xFFFFFFFFull);
        float w = a1[(size_t)sprime * HW + sIdx] * a2[(size_t)qq * HW + qIdx];
        p1 = fmaf(v12, w, p1);
        p2 = fmaf(v21, w, p2);
    }
    r1[tid] = p1;
    r2[tid] = p2;
    __syncthreads();
    for (int sd = 128; sd > 0; sd >>= 1) {     // deterministic tree sum
        if (tid < sd) { r1[tid] += r1[tid + sd]; r2[tid] += r2[tid + sd]; }
        __syncthreads();
    }
    if (tid == 0) {
        out[pair]         = r1[0] * (1.0f / (float)HW);
        out[NPAIR + pair] = r2[0] * (1.0f / (float)HW);
    }
}

// ---------------------------------------------------------------------------
// Launch
// ---------------------------------------------------------------------------
extern "C" void kernel_launch(void* const* d_in, const int* in_sizes, int n_in,
                              void* d_out, int out_size, void* d_ws, size_t ws_size,
                              hipStream_t stream) {
    const float* f1 = (const float*)d_in[0];
    const float* f2 = (const float*)d_in[1];
    const float* W1 = (const float*)d_in[2];
    const float* g1 = (const float*)d_in[3];
    const float* b1 = (const float*)d_in[4];
    const float* m1 = (const float*)d_in[5];
    const float* v1 = (const float*)d_in[6];
    const float* W2 = (const float*)d_in[7];
    const float* g2 = (const float*)d_in[8];
    const float* b2 = (const float*)d_in[9];
    const float* m2 = (const float*)d_in[10];
    const float* v2 = (const float*)d_in[11];
    float* out = (float*)d_out;

    float* ws  = (float*)d_ws;
    v2f* f2p = (v2f*)ws;                                    // 30*256*384 v2f
    v2f* f1p = f2p + (size_t)Q_IMG * CP_DIM * HWP;          // 25*256*384 v2f
    float* a2 = (float*)(f1p + (size_t)S_IMG * CP_DIM * HWP); // 30*361
    float* a1 = a2 + (size_t)Q_IMG * HW;                    // 25*361
    // total ~43.4 MB of workspace

    const int imgBlocks = (S_IMG + Q_IMG) * 3;              // 55 imgs * 3 parts
    k_normalize<<<imgBlocks, 128, 0, stream>>>(f1, f2, f1p, f2p);
    k_meta<<<imgBlocks, 128, 0, stream>>>(f1, f2, W1, g1, b1, m1, v1,
                                          W2, g2, b2, m2, v2, a1, a2);
    k_sim<<<NPAIR, 256, 0, stream>>>(f1p, f2p, a1, a2, out);
}